// SegmentedMaxLoss_39745627357645
// MI455X (gfx1250) — compile-verified
//
#include <hip/hip_runtime.h>

// ---------------- problem constants ----------------
static constexpr int Bn = 32, Tn = 4096, Dn = 128;
static constexpr float ALPHA_C = 0.5f;
static constexpr float NEG_FILL_C = -1e30f;

static constexpr int CHUNKS = 32;                       // blocks per batch
static constexpr int NBLOCKS = Bn * CHUNKS;             // 1024 blocks
static constexpr int ROWS_PER_BLOCK = Tn / CHUNKS;      // 128 rows
static constexpr int WAVES = 8;                         // 256 threads, wave32
static constexpr int ROWS_PER_WAVE = ROWS_PER_BLOCK / WAVES; // 16

typedef __attribute__((ext_vector_type(2))) float v2f;
typedef __attribute__((ext_vector_type(8))) float v8f;

// Monotonic float<->uint key: order-preserving for all finite floats (incl. negatives).
__device__ __forceinline__ unsigned fkey(float f) {
  unsigned u = __float_as_uint(f);
  return (u & 0x80000000u) ? ~u : (u | 0x80000000u);
}
__device__ __forceinline__ float funkey(unsigned k) {
  unsigned u = (k & 0x80000000u) ? (k & 0x7fffffffu) : ~k;
  return __uint_as_float(u);
}

// Branchless dynamic extract from an 8-wide f32 accumulator (lowers to v_cndmask chain).
__device__ __forceinline__ float vsel8(v8f v, int ci) {
  float r = 0.f;
#pragma unroll
  for (int j = 0; j < 8; ++j) r = (ci == j) ? v[j] : r;
  return r;
}

// ---------------- kernel 1: init workspace ----------------
__global__ void segloss_init(unsigned* __restrict__ keys) {
  int i = blockIdx.x * blockDim.x + threadIdx.x;
  if (i < 4 * Bn * Dn) keys[i] = fkey(NEG_FILL_C);
}

// ---------------- kernel 2: streaming pass (BW-bound) ----------------
// keys layout: [0]=maxOutFirst, [1]=maxOutSecond, [2]=maxLabFirst, [3]=maxLabSecond,
// each Bn*Dn uints.  partials: NBLOCKS floats of SSE partial sums.
__global__ void __launch_bounds__(256)
segloss_main(const float* __restrict__ outputs, const float* __restrict__ labels,
             const int* __restrict__ lengths, unsigned* __restrict__ keys,
             float* __restrict__ partials) {
  const int b = blockIdx.x / CHUNKS;
  const int chunk = blockIdx.x % CHUNKS;
  const int tid = threadIdx.x;
  const int wave = tid >> 5, lane = tid & 31;

  const int len = lengths[b];
  const int half = len >> 1;

  const size_t bbase = (size_t)b * Tn * Dn;
  const float4* Ob = (const float4*)(outputs + bbase);
  const float4* Lb = (const float4*)(labels + bbase);
  const int row0 = chunk * ROWS_PER_BLOCK + wave * ROWS_PER_WAVE;

  float4 mof = {NEG_FILL_C, NEG_FILL_C, NEG_FILL_C, NEG_FILL_C};
  float4 mos = mof, mlf = mof, mls = mof;
  float sse = 0.f;

#pragma unroll 4
  for (int i = 0; i < ROWS_PER_WAVE; ++i) {
    const int t = row0 + i;
    const size_t idx = (size_t)t * (Dn / 4) + lane;   // one wave covers a full 128-f row
    const float4 o = Ob[idx];
    const float4 l = Lb[idx];
    if (i + 2 < ROWS_PER_WAVE) {                      // gfx1250 global_prefetch_b8 path
      __builtin_prefetch(&Ob[idx + 2 * (Dn / 4)], 0, 0);
      __builtin_prefetch(&Lb[idx + 2 * (Dn / 4)], 0, 0);
    }
    const float dx = o.x - l.x, dy = o.y - l.y, dz = o.z - l.z, dw = o.w - l.w;
    sse = fmaf(dx, dx, fmaf(dy, dy, fmaf(dz, dz, fmaf(dw, dw, sse))));
    if (t < half) {            // wave-uniform branch (t uniform per wave)
      mof.x = fmaxf(mof.x, o.x); mof.y = fmaxf(mof.y, o.y);
      mof.z = fmaxf(mof.z, o.z); mof.w = fmaxf(mof.w, o.w);
      mlf.x = fmaxf(mlf.x, l.x); mlf.y = fmaxf(mlf.y, l.y);
      mlf.z = fmaxf(mlf.z, l.z); mlf.w = fmaxf(mlf.w, l.w);
    } else if (t < len) {
      mos.x = fmaxf(mos.x, o.x); mos.y = fmaxf(mos.y, o.y);
      mos.z = fmaxf(mos.z, o.z); mos.w = fmaxf(mos.w, o.w);
      mls.x = fmaxf(mls.x, l.x); mls.y = fmaxf(mls.y, l.y);
      mls.z = fmaxf(mls.z, l.z); mls.w = fmaxf(mls.w, l.w);
    }
  }

  // --- SSE: wave32 shfl tree -> LDS -> one fixed-order partial per block ---
#pragma unroll
  for (int off = 16; off; off >>= 1) sse += __shfl_xor(sse, off, 32);
  __shared__ float ssw[WAVES];
  if (lane == 0) ssw[wave] = sse;
  __syncthreads();
  if (tid == 0) {
    float s = 0.f;
#pragma unroll
    for (int w = 0; w < WAVES; ++w) s += ssw[w];
    partials[blockIdx.x] = s;
  }

  // --- maxes: combine 8 waves in LDS, then monotonic-uint atomicMax to L2 ---
  __shared__ float sm[WAVES][Dn];
  const int dbase = lane * 4;
  unsigned* karr[4] = {keys, keys + Bn * Dn, keys + 2 * Bn * Dn, keys + 3 * Bn * Dn};

  auto reduceArr = [&](const float4 v, unsigned* kdst) {
    sm[wave][dbase + 0] = v.x; sm[wave][dbase + 1] = v.y;
    sm[wave][dbase + 2] = v.z; sm[wave][dbase + 3] = v.w;
    __syncthreads();
    if (tid < Dn) {
      float m = sm[0][tid];
#pragma unroll
      for (int w = 1; w < WAVES; ++w) m = fmaxf(m, sm[w][tid]);
      atomicMax(&kdst[b * Dn + tid], fkey(m));
    }
    __syncthreads();
  };
  reduceArr(mof, karr[0]);
  reduceArr(mos, karr[1]);
  reduceArr(mlf, karr[2]);
  reduceArr(mls, karr[3]);
}

// ---------------- kernel 3: finalize (one wave, uses V_WMMA_F32_16X16X4_F32) ---
// S[b] = sum_d d[b,d]^2 computed as the diagonal of the Gram matrix D*D^T via WMMA.
__global__ void __launch_bounds__(32)
segloss_finalize(const unsigned* __restrict__ keys, const float* __restrict__ partials,
                 const int* __restrict__ lengths, float* __restrict__ out) {
  __shared__ float M1[Bn][Dn];   // d1 = segmaxOutFirst - segmaxLabFirst
  __shared__ float M2[Bn][Dn];   // d2
  const int lane = threadIdx.x;  // 32 threads == 1 wave

  const unsigned* kOF = keys;
  const unsigned* kOS = keys + Bn * Dn;
  const unsigned* kLF = keys + 2 * Bn * Dn;
  const unsigned* kLS = keys + 3 * Bn * Dn;

  for (int idx = lane; idx < Bn * Dn; idx += 32) {
    M1[idx >> 7][idx & 127] = funkey(kOF[idx]) - funkey(kLF[idx]);
    M2[idx >> 7][idx & 127] = funkey(kOS[idx]) - funkey(kLS[idx]);
  }
  __syncthreads();

  const bool hi = lane >= 16;
  const int ml = lane & 15;
  // Branchless diagonal-extraction setup:
  // lanes 0-7 hold C[lane,lane] in c[lane]; lanes 24-31 hold C[lane-16,lane-16] in c[lane-24].
  const bool active = (lane < 8) || (lane >= 24);
  const int dm = (lane < 8) ? lane : (lane - 16);   // batch row within group (clamped use below)
  const int ci = (lane < 8) ? lane : (lane - 24);   // accumulator VGPR index
  float tot1 = 0.f, tot2 = 0.f;

  for (int g = 0; g < 2; ++g) {            // two groups of 16 batches
    v8f c1 = {}; v8f c2 = {};
    const float* r1 = M1[g * 16 + ml];
    const float* r2 = M2[g * 16 + ml];
    for (int k0 = 0; k0 < Dn; k0 += 4) {
      // A(16x4) lane layout: lo lanes K={0,1}, hi lanes K={2,3}
      const int kA0 = k0 + (hi ? 2 : 0);
      // B(4x16) lane layout: VGPR0 rows {0,1}, VGPR1 rows {2,3}; hi lanes take odd rows
      const int kB0 = k0 + (hi ? 1 : 0);
      v2f a1 = {r1[kA0], r1[kA0 + 1]};
      v2f b1 = {r1[kB0], r1[kB0 + 2]};
      v2f a2 = {r2[kA0], r2[kA0 + 1]};
      v2f b2 = {r2[kB0], r2[kB0 + 2]};
      c1 = __builtin_amdgcn_wmma_f32_16x16x4_f32(false, a1, false, b1, (short)0, c1, false, false);
      c2 = __builtin_amdgcn_wmma_f32_16x16x4_f32(false, a2, false, b2, (short)0, c2, false, false);
    }
    // Branchless: inactive lanes contribute 0 via weight.
    const int bidx = g * 16 + ((dm >= 0 && dm < 16) ? dm : 0);
    const float w = (active && lengths[bidx] >= 2) ? 1.f : 0.f;
    tot1 += w * vsel8(c1, active ? ci : -1);
    tot2 += w * vsel8(c2, active ? ci : -1);
  }

  // deterministic fixed-order SSE total
  float sse = 0.f;
  for (int i = lane; i < NBLOCKS; i += 32) sse += partials[i];

#pragma unroll
  for (int off = 16; off; off >>= 1) {
    sse  += __shfl_xor(sse,  off, 32);
    tot1 += __shfl_xor(tot1, off, 32);
    tot2 += __shfl_xor(tot2, off, 32);
  }
  if (lane == 0) {
    const float base = sse / (float)((size_t)Bn * Tn * Dn);       // 2^24, exact
    out[0] = base + ALPHA_C * ((tot1 + tot2) / (float)(Dn * Bn)); // /D (mean) and /B
  }
}

// ---------------- launch ----------------
extern "C" void kernel_launch(void* const* d_in, const int* in_sizes, int n_in,
                              void* d_out, int out_size, void* d_ws, size_t ws_size,
                              hipStream_t stream) {
  const float* outputs = (const float*)d_in[0];
  const float* labels  = (const float*)d_in[1];
  const int*   lengths = (const int*)d_in[2];

  unsigned* keys = (unsigned*)d_ws;                                  // 4*Bn*Dn uints (64 KB)
  float* partials = (float*)((char*)d_ws + (size_t)4 * Bn * Dn * sizeof(unsigned)); // NBLOCKS floats

  segloss_init<<<(4 * Bn * Dn + 255) / 256, 256, 0, stream>>>(keys);
  segloss_main<<<NBLOCKS, 256, 0, stream>>>(outputs, labels, lengths, keys, partials);
  segloss_finalize<<<1, 32, 0, stream>>>(keys, partials, lengths, (float*)d_out);
}